// LSTM_GNN_Feedback_11433202942553
// MI455X (gfx1250) — compile-verified
//
#include <hip/hip_runtime.h>
#include <hip/hip_bf16.h>
#include <math.h>

typedef float v2f __attribute__((ext_vector_type(2)));
typedef float v8f __attribute__((ext_vector_type(8)));

#define T_STEPS 10
#define H 64
#define DSF 10
#define DTF 10
#define XF 20
#define KIN 74     /* DT + H */
#define FUSED 128
#define EC 64
#define WPB 8      /* waves per block (256 threads, wave32) */

// ---------------------------------------------------------------------------
// WMMA helpers: D = A(16x4) * B(4x16) + C, fp32, CDNA5 V_WMMA_F32_16X16X4_F32
// ---------------------------------------------------------------------------
__device__ __forceinline__ v8f wmma4(v2f a, v2f b, v8f c) {
  return __builtin_amdgcn_wmma_f32_16x16x4_f32(false, a, false, b, (short)0, c,
                                               false, false);
}

// A fragment: rows row0..row0+15, cols k0..k0+3 of a row-major [*, ld] matrix.
// ISA layout: lanes 0-15 hold M=lane, VGPR0=K0,VGPR1=K1; lanes 16-31: K2,K3.
__device__ __forceinline__ v2f a_frag(const float* __restrict__ base, int ld,
                                      int row0, int k0, int lane) {
  const int m = lane & 15;
  const int k = k0 + ((lane >> 4) << 1);
  const float* p = base + (size_t)(row0 + m) * ld + k;
  v2f r; r.x = p[0]; r.y = p[1]; return r;
}

__device__ __forceinline__ v2f a_frag_lds(const float* base, int ld, int k0,
                                          int lane) {
  const int m = lane & 15;
  const int k = k0 + ((lane >> 4) << 1);
  const float* p = base + m * ld + k;
  v2f r; r.x = p[0]; r.y = p[1]; return r;
}

// B fragment for W^T where W is row-major [out x ldw]: B[k][n] = W[n][k].
__device__ __forceinline__ v2f bT_frag(const float* __restrict__ W, int ldw,
                                       int n0, int k0, int lane) {
  const int n = n0 + (lane & 15);
  const int k = k0 + ((lane >> 4) << 1);
  const float* p = W + (size_t)n * ldw + k;
  v2f r; r.x = p[0]; r.y = p[1]; return r;
}

// Guarded variant for ragged K (K=74).
__device__ __forceinline__ v2f bT_frag_g(const float* __restrict__ W, int ldw,
                                         int n0, int k0, int kmax, int lane) {
  const int n = n0 + (lane & 15);
  const int k = k0 + ((lane >> 4) << 1);
  v2f r;
  r.x = (k     < kmax) ? W[(size_t)n * ldw + k    ] : 0.0f;
  r.y = (k + 1 < kmax) ? W[(size_t)n * ldw + k + 1] : 0.0f;
  return r;
}

// A fragment of inp = [x_t (10) | g (64)], K padded 74 -> 76 with zeros.
__device__ __forceinline__ float inp_elem(const float* __restrict__ x,
                                          const float* gl, int node, int m,
                                          int k, int t) {
  if (k < DTF) return x[(size_t)node * (XF * T_STEPS) + (size_t)(DSF + k) * T_STEPS + t];
  if (k < KIN) return gl[m * H + (k - DTF)];
  return 0.0f;
}
__device__ __forceinline__ v2f inp_frag(const float* __restrict__ x,
                                        const float* gl, int row0, int k0,
                                        int t, int lane) {
  const int m = lane & 15;
  const int k = k0 + ((lane >> 4) << 1);
  const int node = row0 + m;
  v2f r;
  r.x = inp_elem(x, gl, node, m, k,     t);
  r.y = inp_elem(x, gl, node, m, k + 1, t);
  return r;
}

__device__ __forceinline__ float sigmoidf(float v) {
  return 1.0f / (1.0f + __expf(-v));
}

// ---------------------------------------------------------------------------
// Utility kernels
// ---------------------------------------------------------------------------
__global__ void zero_kernel(float* __restrict__ p, long long n) {
  const long long i = (long long)blockIdx.x * blockDim.x + threadIdx.x;
  if (i < n) p[i] = 0.0f;
}

// agg[dst] += h[src] : 16 threads/edge, each does a float4 gather + 4 atomics.
// h and agg are L2-resident (25.6 MB each vs 192 MB L2) -> on-chip atomics.
__global__ void scatter_kernel(const float* __restrict__ h,
                               const int* __restrict__ src,
                               const int* __restrict__ dst,
                               float* __restrict__ agg, int E) {
  const long long idx = (long long)blockIdx.x * blockDim.x + threadIdx.x;
  const int e = (int)(idx >> 4);
  if (e >= E) return;
  const int c4 = ((int)idx & 15) << 2;
  const int s = src[e];
  const int d = dst[e];
  const float4 v = *(const float4*)(h + (size_t)s * H + c4);
  float* ap = agg + (size_t)d * H + c4;
  atomicAdd(ap + 0, v.x);
  atomicAdd(ap + 1, v.y);
  atomicAdd(ap + 2, v.z);
  atomicAdd(ap + 3, v.w);
}

// ---------------------------------------------------------------------------
// Fused per-step kernel: gconv + LSTM gates + cell update.
// One wave = one 16-node tile. All WMMA under wave-uniform guards (EXEC full).
// A-panels are register-blocked (v2f[16]) and reused across all column tiles.
// ---------------------------------------------------------------------------
__global__ void fused_step_kernel(const float* __restrict__ x,
                                  const float* __restrict__ agg,
                                  const float* __restrict__ h_in,
                                  float* __restrict__ h_out,
                                  float* __restrict__ c_st,
                                  float* __restrict__ g_out,
                                  const float* __restrict__ W_rel,
                                  const float* __restrict__ b_rel,
                                  const float* __restrict__ W_root,
                                  const float* __restrict__ W_ih,
                                  const float* __restrict__ W_hh,
                                  const float* __restrict__ b_ih,
                                  const float* __restrict__ b_hh,
                                  int t, int N, int write_g) {
  __shared__ float g_lds[WPB][16 * H];
  const int wave = threadIdx.x >> 5;
  const int lane = threadIdx.x & 31;
  const int row0 = (blockIdx.x * WPB + wave) * 16;
  const bool active = row0 < N;
  float* gl = g_lds[wave];
  const int colL = lane & 15;

  v2f aH[16];  // h row-panel fragments, loaded once, reused in both phases

  // Phase 1: g = agg @ W_rel^T + b_rel + h @ W_root^T   (16x64 tile)
  if (active) {
    v2f aAgg[16];
#pragma unroll
    for (int kk = 0; kk < 16; ++kk) {
      aAgg[kk] = a_frag(agg,  H, row0, kk * 4, lane);
      aH[kk]   = a_frag(h_in, H, row0, kk * 4, lane);
    }
    for (int nt = 0; nt < 4; ++nt) {
      const int n0 = nt * 16;
      v8f acc;
      const float bias = b_rel[n0 + colL];
#pragma unroll
      for (int r = 0; r < 8; ++r) acc[r] = bias;
#pragma unroll
      for (int kk = 0; kk < 16; ++kk) {
        acc = wmma4(aAgg[kk], bT_frag(W_rel,  H, n0, kk * 4, lane), acc);
        acc = wmma4(aH[kk],   bT_frag(W_root, H, n0, kk * 4, lane), acc);
      }
#pragma unroll
      for (int r = 0; r < 8; ++r) {
        const int m = (lane < 16) ? r : (r + 8);
        gl[m * H + n0 + colL] = acc[r];
        if (write_g) g_out[(size_t)(row0 + m) * H + n0 + colL] = acc[r];
      }
    }
  }
  __syncthreads();
  if (!active) return;

  // Phase 2: gates = [x_t|g] @ W_ih^T + b_ih + h @ W_hh^T + b_hh, then cell.
  // Gate layout in N-dim: [i(0:64) | f(64:128) | g(128:192) | o(192:256)].
  v2f aInp[19];
#pragma unroll
  for (int kk = 0; kk < 19; ++kk)
    aInp[kk] = inp_frag(x, gl, row0, kk * 4, t, lane);

  for (int jt = 0; jt < 4; ++jt) {
    const int n0 = jt * 16;
    v8f ai, af, ag, ao;
    const float bi  = b_ih[n0 + colL]         + b_hh[n0 + colL];
    const float bf  = b_ih[H + n0 + colL]     + b_hh[H + n0 + colL];
    const float bg  = b_ih[2 * H + n0 + colL] + b_hh[2 * H + n0 + colL];
    const float bo_ = b_ih[3 * H + n0 + colL] + b_hh[3 * H + n0 + colL];
#pragma unroll
    for (int r = 0; r < 8; ++r) { ai[r] = bi; af[r] = bf; ag[r] = bg; ao[r] = bo_; }

#pragma unroll
    for (int kk = 0; kk < 19; ++kk) {  // K = 74, zero-padded to 76
      const int k = kk * 4;
      ai = wmma4(aInp[kk], bT_frag_g(W_ih, KIN, n0,         k, KIN, lane), ai);
      af = wmma4(aInp[kk], bT_frag_g(W_ih, KIN, H + n0,     k, KIN, lane), af);
      ag = wmma4(aInp[kk], bT_frag_g(W_ih, KIN, 2 * H + n0, k, KIN, lane), ag);
      ao = wmma4(aInp[kk], bT_frag_g(W_ih, KIN, 3 * H + n0, k, KIN, lane), ao);
    }
#pragma unroll
    for (int kk = 0; kk < 16; ++kk) {
      const int k = kk * 4;
      ai = wmma4(aH[kk], bT_frag(W_hh, H, n0,         k, lane), ai);
      af = wmma4(aH[kk], bT_frag(W_hh, H, H + n0,     k, lane), af);
      ag = wmma4(aH[kk], bT_frag(W_hh, H, 2 * H + n0, k, lane), ag);
      ao = wmma4(aH[kk], bT_frag(W_hh, H, 3 * H + n0, k, lane), ao);
    }
#pragma unroll
    for (int r = 0; r < 8; ++r) {
      const int m = (lane < 16) ? r : (r + 8);
      const size_t idx = (size_t)(row0 + m) * H + n0 + colL;
      const float cp = c_st[idx];
      const float iv = sigmoidf(ai[r]);
      const float fv = sigmoidf(af[r]);
      const float gv = tanhf(ag[r]);
      const float ov = sigmoidf(ao[r]);
      const float c2 = fv * cp + iv * gv;
      c_st[idx]  = c2;                 // in-place: single read-then-write/lane
      h_out[idx] = ov * tanhf(c2);     // double-buffered
    }
  }
}

// ---------------------------------------------------------------------------
// LayerNorm over fused = [h | g] (128 ch). One wave per node, shfl reduction.
// ---------------------------------------------------------------------------
__global__ void layernorm_kernel(const float* __restrict__ h,
                                 const float* __restrict__ g,
                                 const float* __restrict__ gamma,
                                 const float* __restrict__ beta,
                                 float* __restrict__ normed, int N) {
  const int wave = threadIdx.x >> 5;
  const int lane = threadIdx.x & 31;
  const int node = blockIdx.x * 8 + wave;
  if (node >= N) return;
  float v[4];
  const int j0 = lane * 4;
#pragma unroll
  for (int q = 0; q < 4; ++q) {
    const int j = j0 + q;
    v[q] = (j < H) ? h[(size_t)node * H + j] : g[(size_t)node * H + (j - H)];
  }
  float s  = v[0] + v[1] + v[2] + v[3];
  float s2 = v[0] * v[0] + v[1] * v[1] + v[2] * v[2] + v[3] * v[3];
#pragma unroll
  for (int m = 16; m >= 1; m >>= 1) {
    s  += __shfl_xor(s,  m, 32);
    s2 += __shfl_xor(s2, m, 32);
  }
  const float mean = s * (1.0f / FUSED);
  const float var  = s2 * (1.0f / FUSED) - mean * mean;
  const float inv  = rsqrtf(var + 1e-5f);
#pragma unroll
  for (int q = 0; q < 4; ++q) {
    const int j = j0 + q;
    normed[(size_t)node * FUSED + j] = (v[q] - mean) * inv * gamma[j] + beta[j];
  }
}

// ---------------------------------------------------------------------------
// MLP head: a1 = relu(normed@W1^T+b1); hidden = relu(a1@W2^T+b2);
// logits = hidden@Wo^T+bo.  out = [logits | logits | hidden].
// ---------------------------------------------------------------------------
__global__ void head_kernel(const float* __restrict__ normed,
                            const float* __restrict__ W1, const float* __restrict__ b1,
                            const float* __restrict__ W2, const float* __restrict__ b2,
                            const float* __restrict__ Wo, const float* __restrict__ bo,
                            float* __restrict__ out, int N) {
  __shared__ float a1_lds[WPB][16 * EC];
  const int wave = threadIdx.x >> 5;
  const int lane = threadIdx.x & 31;
  const int row0 = (blockIdx.x * WPB + wave) * 16;
  const bool active = row0 < N;
  float* al = a1_lds[wave];
  const int colL = lane & 15;

  if (active) {
    v2f aN[32];  // normed row-panel (K=128), reused across 4 column tiles
#pragma unroll
    for (int kk = 0; kk < 32; ++kk)
      aN[kk] = a_frag(normed, FUSED, row0, kk * 4, lane);
    for (int nt = 0; nt < 4; ++nt) {
      const int n0 = nt * 16;
      v8f acc;
      const float bias = b1[n0 + colL];
#pragma unroll
      for (int r = 0; r < 8; ++r) acc[r] = bias;
#pragma unroll
      for (int kk = 0; kk < 32; ++kk)
        acc = wmma4(aN[kk], bT_frag(W1, FUSED, n0, kk * 4, lane), acc);
#pragma unroll
      for (int r = 0; r < 8; ++r) {
        const int m = (lane < 16) ? r : (r + 8);
        al[m * EC + n0 + colL] = fmaxf(acc[r], 0.0f);
      }
    }
  }
  __syncthreads();
  if (!active) return;

  v2f aA[16];  // a1 row-panel from LDS (K=64), reused across 8 column tiles
#pragma unroll
  for (int kk = 0; kk < 16; ++kk)
    aA[kk] = a_frag_lds(al, EC, kk * 4, lane);

  float part[8];
#pragma unroll
  for (int r = 0; r < 8; ++r) part[r] = 0.0f;

  for (int nt = 0; nt < 8; ++nt) {
    const int n0 = nt * 16;
    v8f acc;
    const float bias = b2[n0 + colL];
#pragma unroll
    for (int r = 0; r < 8; ++r) acc[r] = bias;
#pragma unroll
    for (int kk = 0; kk < 16; ++kk)
      acc = wmma4(aA[kk], bT_frag(W2, EC, n0, kk * 4, lane), acc);
    const float wo = Wo[n0 + colL];
#pragma unroll
    for (int r = 0; r < 8; ++r) {
      const int m = (lane < 16) ? r : (r + 8);
      const float hv = fmaxf(acc[r], 0.0f);
      out[(size_t)2 * N + (size_t)(row0 + m) * FUSED + n0 + colL] = hv;
      part[r] += hv * wo;
    }
  }
#pragma unroll
  for (int m = 8; m >= 1; m >>= 1) {
#pragma unroll
    for (int r = 0; r < 8; ++r) part[r] += __shfl_xor(part[r], m, 32);
  }
  if (lane == 0 || lane == 16) {
    const int base = (lane == 0) ? 0 : 8;
    const float bb = bo[0];
#pragma unroll
    for (int r = 0; r < 8; ++r) {
      const int node = row0 + base + r;
      const float lv = part[r] + bb;
      out[node]     = lv;   // output
      out[N + node] = lv;   // logits
    }
  }
}

// ---------------------------------------------------------------------------
extern "C" void kernel_launch(void* const* d_in, const int* in_sizes, int n_in,
                              void* d_out, int out_size, void* d_ws,
                              size_t ws_size, hipStream_t stream) {
  (void)n_in; (void)out_size; (void)ws_size;
  const float* x      = (const float*)d_in[0];
  const int*   ei     = (const int*)d_in[1];
  // d_in[2]/d_in[3] (W_static/b_static) contribute 0*static_embed -> unused.
  const float* W_ih   = (const float*)d_in[4];
  const float* W_hh   = (const float*)d_in[5];
  const float* b_ih   = (const float*)d_in[6];
  const float* b_hh   = (const float*)d_in[7];
  const float* W_rel  = (const float*)d_in[8];
  const float* b_rel  = (const float*)d_in[9];
  const float* W_root = (const float*)d_in[10];
  const float* gamma  = (const float*)d_in[11];
  const float* beta   = (const float*)d_in[12];
  const float* W1     = (const float*)d_in[13];
  const float* b1     = (const float*)d_in[14];
  const float* W2     = (const float*)d_in[15];
  const float* b2     = (const float*)d_in[16];
  const float* Wo     = (const float*)d_in[17];
  const float* bo     = (const float*)d_in[18];

  const int N = in_sizes[0] / (XF * T_STEPS);
  const int E = in_sizes[1] / 2;
  const int* src = ei;
  const int* dst = ei + E;

  const long long NH = (long long)N * H;
  float* ws  = (float*)d_ws;
  float* hb0 = ws;                 // h double buffer A  (N*64)
  float* hb1 = ws + NH;            // h double buffer B  (N*64)
  float* cb  = ws + 2 * NH;        // c (in-place)       (N*64)
  float* agg = ws + 3 * NH;        // segment sums       (N*64)
  float* gb  = ws + 4 * NH;        // final gconv output (N*64)
  float* nm  = ws + 5 * NH;        // layernorm output   (N*128)
  float* out = (float*)d_out;

  const int tiles       = (N + 15) / 16;
  const int gemm_blocks = (tiles + WPB - 1) / WPB;
  const int zb          = (int)((NH + 255) / 256);
  const long long sthreads = (long long)E * 16;
  const int sb          = (int)((sthreads + 255) / 256);
  const int lnb         = (N + 7) / 8;

  zero_kernel<<<zb, 256, 0, stream>>>(hb0, NH);  // h0 = 0
  zero_kernel<<<zb, 256, 0, stream>>>(cb, NH);   // c0 = 0

  int cur = 0;
  for (int t = 0; t < T_STEPS; ++t) {
    float* hin  = cur ? hb1 : hb0;
    float* hout = cur ? hb0 : hb1;
    zero_kernel<<<zb, 256, 0, stream>>>(agg, NH);
    scatter_kernel<<<sb, 256, 0, stream>>>(hin, src, dst, agg, E);
    fused_step_kernel<<<gemm_blocks, WPB * 32, 0, stream>>>(
        x, agg, hin, hout, cb, gb, W_rel, b_rel, W_root, W_ih, W_hh, b_ih,
        b_hh, t, N, (t == T_STEPS - 1) ? 1 : 0);
    cur ^= 1;
  }
  float* hfin = cur ? hb1 : hb0;
  layernorm_kernel<<<lnb, 256, 0, stream>>>(hfin, gb, gamma, beta, nm, N);
  head_kernel<<<gemm_blocks, WPB * 32, 0, stream>>>(nm, W1, b1, W2, b2, Wo, bo,
                                                    out, N);
}